// GAT_30442728194257
// MI455X (gfx1250) — compile-verified
//
#include <hip/hip_runtime.h>
#include <hip/hip_bf16.h>

typedef __attribute__((ext_vector_type(2))) float v2f;
typedef __attribute__((ext_vector_type(8))) float v8f;

#define SLOPE 0.2f

static __host__ __device__ inline int cdiv(int a, int b) { return (a + b - 1) / b; }

// Native f32 atomic add (GLOBAL_ATOMIC_ADD_F32), avoids CAS-loop lowering.
__device__ inline void atomicAddF32(float* p, float v) { unsafeAtomicAdd(p, v); }

// ---------------------------------------------------------------------------
// Fill kernel (d_ws is poisoned; init everything read-before-write)
// ---------------------------------------------------------------------------
__global__ void gat_fill_f32(float* __restrict__ p, float v, int n) {
    int i = blockIdx.x * blockDim.x + threadIdx.x;
    if (i < n) p[i] = v;
}

// ---------------------------------------------------------------------------
// GEMM1: H = feat @ W1   [N,256] x [256,256] -> [N,256], fp32 WMMA 16x16x4
// Block: 256 threads = 8 waves, computes 32 rows x 256 cols.
// LDS: 32x256 A panel (32KB) + 32x256 W k-chunk (32KB), both loaded b128.
// Wave w owns cols [32w,32w+32) as two 16x16 col-tiles x two row-tiles
// -> 4 accumulators, 4 WMMAs per K=4 step (B fragments amortized 2x).
// ---------------------------------------------------------------------------
__global__ void gat_gemm1_wmma(const float* __restrict__ A,   // [N,256]
                               const float* __restrict__ W,   // [256,256]
                               float* __restrict__ H,         // [N,256]
                               int n_nodes) {
    __shared__ float As[32 * 256];   // 32 KB
    __shared__ float Ws[32 * 256];   // 32 KB

    const int row0 = blockIdx.x * 32;
    const int tid  = threadIdx.x;
    const int wave = tid >> 5;
    const int lane = tid & 31;

    // Cooperative A-panel load, float4-coalesced (clamp rows: EXEC stays full)
    for (int i = tid; i < 32 * 64; i += 256) {       // i indexes a float4
        int r  = i >> 6;
        int c4 = (i & 63) << 2;
        int rr = row0 + r; if (rr >= n_nodes) rr = n_nodes - 1;
        *(float4*)(&As[r * 256 + c4]) = *(const float4*)(A + (size_t)rr * 256 + c4);
    }

    const int m   = lane & 15;           // row within A tile / col within B tile
    const int kk  = (lane >> 4) << 1;    // 0 or 2 : K sub-pair for this half-wave
    const int n0  = wave * 32;
    const int n1  = n0 + 16;

    v8f c00 = {}, c01 = {}, c10 = {}, c11 = {};   // [rowtile][coltile]

    for (int kc = 0; kc < 256; kc += 32) {
        __syncthreads();   // As ready (first iter) / prior Ws reads done (later iters)
        // Cooperative W-chunk load: rows [kc, kc+32) x 256 cols, float4-coalesced
        for (int i = tid; i < 32 * 64; i += 256) {
            int r  = i >> 6;
            int c4 = (i & 63) << 2;
            *(float4*)(&Ws[r * 256 + c4]) = *(const float4*)(W + (size_t)(kc + r) * 256 + c4);
        }
        __syncthreads();

#pragma unroll
        for (int k = 0; k < 32; k += 4) {
            v2f a0, a1, b0, b1;
            a0.x = As[m * 256 + kc + k + kk];
            a0.y = As[m * 256 + kc + k + kk + 1];
            a1.x = As[(m + 16) * 256 + kc + k + kk];
            a1.y = As[(m + 16) * 256 + kc + k + kk + 1];
            b0.x = Ws[(k + kk) * 256 + n0 + m];
            b0.y = Ws[(k + kk + 1) * 256 + n0 + m];
            b1.x = Ws[(k + kk) * 256 + n1 + m];
            b1.y = Ws[(k + kk + 1) * 256 + n1 + m];
            c00 = __builtin_amdgcn_wmma_f32_16x16x4_f32(false, a0, false, b0, (short)0, c00, false, false);
            c01 = __builtin_amdgcn_wmma_f32_16x16x4_f32(false, a0, false, b1, (short)0, c01, false, false);
            c10 = __builtin_amdgcn_wmma_f32_16x16x4_f32(false, a1, false, b0, (short)0, c10, false, false);
            c11 = __builtin_amdgcn_wmma_f32_16x16x4_f32(false, a1, false, b1, (short)0, c11, false, false);
        }
    }

    // C/D layout: VGPR v, lanes 0-15 -> (M=v, N=m); lanes 16-31 -> (M=v+8, N=m)
    const int mhi = (lane >> 4) * 8;
    for (int v = 0; v < 8; ++v) {
        int r0 = row0 + mhi + v;
        int r1 = r0 + 16;
        if (r0 < n_nodes) {
            H[(size_t)r0 * 256 + n0 + m] = c00[v];
            H[(size_t)r0 * 256 + n1 + m] = c01[v];
        }
        if (r1 < n_nodes) {
            H[(size_t)r1 * 256 + n0 + m] = c10[v];
            H[(size_t)r1 * 256 + n1 + m] = c11[v];
        }
    }
}

// ---------------------------------------------------------------------------
// el/er per (node, head): 32-wide dots against a_l / a_r
// ---------------------------------------------------------------------------
__global__ void gat_attn_scores(const float* __restrict__ H,
                                const float* __restrict__ al,
                                const float* __restrict__ ar,
                                float* __restrict__ el, float* __restrict__ er,
                                int n_nodes) {
    int i = blockIdx.x * blockDim.x + threadIdx.x;   // n*8 + h
    if (i >= n_nodes * 8) return;
    int n = i >> 3, h = i & 7;
    const float* hp  = H + (size_t)n * 256 + h * 32;
    const float* alp = al + h * 32;
    const float* arp = ar + h * 32;
    float sl = 0.f, sr = 0.f;
    for (int d = 0; d < 32; ++d) { float v = hp[d]; sl += v * alp[d]; sr += v * arp[d]; }
    el[i] = sl; er[i] = sr;
}

// ---------------------------------------------------------------------------
// Atomic float max via int/uint ordering trick (GLOBAL_ATOMIC_{MAX_I32,MIN_U32})
// ---------------------------------------------------------------------------
__device__ inline void atomicMaxFloat(float* addr, float value) {
    if (value >= 0.0f)
        atomicMax((int*)addr, __float_as_int(value));
    else
        atomicMin((unsigned int*)addr, __float_as_uint(value));
}

// e = leakyrelu(el[src] + er[dst]) ; m[dst] = segmax(e)      (generic in #heads)
__global__ void gat_edge_e_max(const int* __restrict__ src, const int* __restrict__ dst,
                               const float* __restrict__ el, const float* __restrict__ er,
                               float* __restrict__ e, float* __restrict__ m,
                               int n_edges, int heads) {
    int i = blockIdx.x * blockDim.x + threadIdx.x;
    if (i >= n_edges * heads) return;
    int ed = i / heads, h = i - ed * heads;
    float v = el[src[ed] * heads + h] + er[dst[ed] * heads + h];
    v = (v >= 0.f) ? v : SLOPE * v;
    e[i] = v;
    atomicMaxFloat(&m[dst[ed] * heads + h], v);
}

// ue = exp(e - m[dst]) (in place) ; s[dst] = segsum(ue)
__global__ void gat_edge_expsum(const int* __restrict__ dst,
                                float* __restrict__ e, const float* __restrict__ m,
                                float* __restrict__ s, int n_edges, int heads) {
    int i = blockIdx.x * blockDim.x + threadIdx.x;
    if (i >= n_edges * heads) return;
    int ed = i / heads, h = i - ed * heads;
    float v = expf(e[i] - m[dst[ed] * heads + h]);
    e[i] = v;
    atomicAddF32(&s[dst[ed] * heads + h], v);
}

// agg[dst, c] += h[src, c] * ue/s[dst]   — one thread per (edge, feature col)
__global__ void gat_aggregate(const int* __restrict__ src, const int* __restrict__ dst,
                              const float* __restrict__ H, const float* __restrict__ ue,
                              const float* __restrict__ s, float* __restrict__ agg,
                              int n_edges, int heads, int dim) {
    int hd = heads * dim;
    int i = blockIdx.x * blockDim.x + threadIdx.x;
    if (i >= n_edges * hd) return;
    int ed = i / hd, c = i - ed * hd;
    int h = c / dim;
    int sn = src[ed], dn = dst[ed];
    float alpha = ue[ed * heads + h] / s[dn * heads + h];
    atomicAddF32(&agg[(size_t)dn * hd + c], H[(size_t)sn * hd + c] * alpha);
}

// x = elu(x + b[c % HD])  (in place)
__global__ void gat_elu_bias(float* __restrict__ x, const float* __restrict__ b,
                             int total, int hd) {
    int i = blockIdx.x * blockDim.x + threadIdx.x;
    if (i >= total) return;
    float v = x[i] + b[i % hd];
    x[i] = (v > 0.f) ? v : expm1f(v);
}

// Layer-2 linear: h2 = x @ W2 ([N,256]x[256,3]) fused with el2/er2
__global__ void gat_l2_linear(const float* __restrict__ x, const float* __restrict__ W2,
                              const float* __restrict__ al2, const float* __restrict__ ar2,
                              float* __restrict__ h2, float* __restrict__ el2,
                              float* __restrict__ er2, int n_nodes) {
    int n = blockIdx.x * blockDim.x + threadIdx.x;
    if (n >= n_nodes) return;
    float a0 = 0.f, a1 = 0.f, a2 = 0.f;
    const float* xp = x + (size_t)n * 256;
    for (int k = 0; k < 256; ++k) {
        float xv = xp[k];
        a0 += xv * W2[k * 3 + 0];
        a1 += xv * W2[k * 3 + 1];
        a2 += xv * W2[k * 3 + 2];
    }
    h2[n * 3 + 0] = a0; h2[n * 3 + 1] = a1; h2[n * 3 + 2] = a2;
    el2[n] = a0 * al2[0] + a1 * al2[1] + a2 * al2[2];
    er2[n] = a0 * ar2[0] + a1 * ar2[1] + a2 * ar2[2];
}

// out = agg2 + b2  (H2 = 1 head -> mean over heads is identity)
__global__ void gat_finalize(const float* __restrict__ agg2, const float* __restrict__ b2,
                             float* __restrict__ out, int total) {
    int i = blockIdx.x * blockDim.x + threadIdx.x;
    if (i >= total) return;
    out[i] = agg2[i] + b2[i % 3];
}

// ---------------------------------------------------------------------------
extern "C" void kernel_launch(void* const* d_in, const int* in_sizes, int n_in,
                              void* d_out, int out_size, void* d_ws, size_t ws_size,
                              hipStream_t stream) {
    const float* feat = (const float*)d_in[0];
    const int*   src  = (const int*)  d_in[1];
    const int*   dst  = (const int*)  d_in[2];
    const float* W1   = (const float*)d_in[3];
    const float* al1  = (const float*)d_in[4];
    const float* ar1  = (const float*)d_in[5];
    const float* b1   = (const float*)d_in[6];
    const float* W2   = (const float*)d_in[7];
    const float* al2  = (const float*)d_in[8];
    const float* ar2  = (const float*)d_in[9];
    const float* b2   = (const float*)d_in[10];
    float* out = (float*)d_out;

    const int n_nodes = in_sizes[0] / 256;
    const int n_edges = in_sizes[1];

    // Carve scratch from d_ws (256B-aligned slices)
    char* cur = (char*)d_ws;
    auto alloc = [&](size_t bytes) -> float* {
        float* p = (float*)cur;
        cur += (bytes + 255) & ~(size_t)255;
        return p;
    };
    float* H1   = alloc((size_t)n_nodes * 256 * 4);  // layer-1 features h
    float* el1  = alloc((size_t)n_nodes * 8 * 4);
    float* er1  = alloc((size_t)n_nodes * 8 * 4);
    float* m1   = alloc((size_t)n_nodes * 8 * 4);
    float* s1   = alloc((size_t)n_nodes * 8 * 4);
    float* e1   = alloc((size_t)n_edges * 8 * 4);    // e then ue (in place)
    float* agg1 = alloc((size_t)n_nodes * 256 * 4);  // agg then elu'd h (in place)
    float* h2   = alloc((size_t)n_nodes * 3 * 4);
    float* el2  = alloc((size_t)n_nodes * 4);
    float* er2  = alloc((size_t)n_nodes * 4);
    float* m2   = alloc((size_t)n_nodes * 4);
    float* s2   = alloc((size_t)n_nodes * 4);
    float* e2   = alloc((size_t)n_edges * 4);
    float* agg2 = alloc((size_t)n_nodes * 3 * 4);

    const int B = 256;
    const float NEG_INF = -__builtin_huge_valf();

    // ---- Layer 1 ----
    gat_gemm1_wmma<<<cdiv(n_nodes, 32), 256, 0, stream>>>(feat, W1, H1, n_nodes);
    gat_attn_scores<<<cdiv(n_nodes * 8, B), B, 0, stream>>>(H1, al1, ar1, el1, er1, n_nodes);

    gat_fill_f32<<<cdiv(n_nodes * 8, B), B, 0, stream>>>(m1, NEG_INF, n_nodes * 8);
    gat_fill_f32<<<cdiv(n_nodes * 8, B), B, 0, stream>>>(s1, 0.f, n_nodes * 8);
    gat_fill_f32<<<cdiv(n_nodes * 256, B), B, 0, stream>>>(agg1, 0.f, n_nodes * 256);

    gat_edge_e_max<<<cdiv(n_edges * 8, B), B, 0, stream>>>(src, dst, el1, er1, e1, m1, n_edges, 8);
    gat_edge_expsum<<<cdiv(n_edges * 8, B), B, 0, stream>>>(dst, e1, m1, s1, n_edges, 8);
    gat_aggregate<<<cdiv(n_edges * 256, B), B, 0, stream>>>(src, dst, H1, e1, s1, agg1,
                                                            n_edges, 8, 32);
    gat_elu_bias<<<cdiv(n_nodes * 256, B), B, 0, stream>>>(agg1, b1, n_nodes * 256, 256);

    // ---- Layer 2 ----
    gat_l2_linear<<<cdiv(n_nodes, 128), 128, 0, stream>>>(agg1, W2, al2, ar2, h2, el2, er2, n_nodes);

    gat_fill_f32<<<cdiv(n_nodes, B), B, 0, stream>>>(m2, NEG_INF, n_nodes);
    gat_fill_f32<<<cdiv(n_nodes, B), B, 0, stream>>>(s2, 0.f, n_nodes);
    gat_fill_f32<<<cdiv(n_nodes * 3, B), B, 0, stream>>>(agg2, 0.f, n_nodes * 3);

    gat_edge_e_max<<<cdiv(n_edges, B), B, 0, stream>>>(src, dst, el2, er2, e2, m2, n_edges, 1);
    gat_edge_expsum<<<cdiv(n_edges, B), B, 0, stream>>>(dst, e2, m2, s2, n_edges, 1);
    gat_aggregate<<<cdiv(n_edges * 3, B), B, 0, stream>>>(src, dst, h2, e2, s2, agg2,
                                                          n_edges, 1, 3);
    gat_finalize<<<cdiv(n_nodes * 3, B), B, 0, stream>>>(agg2, b2, out, n_nodes * 3);
}